// Capsule_34342558498916
// MI455X (gfx1250) — compile-verified
//
#include <hip/hip_runtime.h>

// Capsule routing for MI455X (gfx1250).
// u_vecs: (B=64, C=64, U=1024) f32 ; W: (U=1024, N=32, C=64, D=16) f32
// out: (B=64, D=16, N=32) f32
//
// Roofline: W(128MB)+u_vecs(16MB) HBM reads ~6.2us @23.3TB/s; u_hat (128MB)
// lives in d_ws and stays L2-resident (192MB L2) for the 3 routing passes.
// fp32 WMMA (v_wmma_f32_16x16x4_f32) keeps reference precision and roughly
// matches the bandwidth floor; lower precision would save no memory traffic.

typedef __attribute__((ext_vector_type(2))) float v2f;
typedef __attribute__((ext_vector_type(4))) float v4f;
typedef __attribute__((ext_vector_type(8))) float v8f;

#define B_DIM 64
#define C_DIM 64
#define U_DIM 1024
#define N_CAP 32
#define D_CAP 16

// ---------------------------------------------------------------------------
// Kernel 0: transpose u_vecs (bc, u) -> uT (u, bc) so per-u A matrices are
// contiguous 16KB blocks (native layout would force stride-4KB gathers).
// ---------------------------------------------------------------------------
__global__ void caps_transpose(const float* __restrict__ in,   // [4096][1024]
                               float* __restrict__ outT) {     // [1024][4096]
  __shared__ float tile[32][33];
  const int u0 = blockIdx.x * 32;
  const int r0 = blockIdx.y * 32;          // bc tile
  const int tx = threadIdx.x & 31;
  const int ty = threadIdx.x >> 5;         // 0..7
  for (int i = 0; i < 32; i += 8)
    tile[ty + i][tx] = in[(size_t)(r0 + ty + i) * U_DIM + u0 + tx];
  __syncthreads();
  for (int i = 0; i < 32; i += 8)
    outT[(size_t)(u0 + ty + i) * (B_DIM * C_DIM) + r0 + tx] = tile[tx][ty + i];
}

// ---------------------------------------------------------------------------
// Kernel 1: per-u GEMM  u_hat[b, n, u, d] = sum_c uT[u, b, c] * W[u, n, c, d]
// One workgroup (16 waves, 512 threads) per u: M=64(b) x K=64(c) x N=512(n*16+d).
// A in LDS (row pitch 68 -> conflict-free, float4-aligned).
// Each wave owns 2 capsules n (32 cols x 64 rows = 8 acc tiles = 64 VGPRs,
// leaving headroom so the 16-step unrolled k-loop does NOT spill).
// Each W byte is read from global exactly once, 64B-coalesced per half-wave.
// ---------------------------------------------------------------------------
__global__ void __launch_bounds__(512, 1)
caps_uhat(const float* __restrict__ uT,   // [U][B*C]
          const float* __restrict__ W,    // [U][N][C][D]
          float* __restrict__ uhat) {     // [B][N][U][D]
  const int u = blockIdx.x;
  __shared__ float As[B_DIM * 68];                        // A[b][c], pitch 68

  // Cooperative load of A (4096 floats) as float4s.
  const float* Ag = uT + (size_t)u * (B_DIM * C_DIM);
  for (int i = threadIdx.x; i < (B_DIM * C_DIM) / 4; i += 512) {
    const int e = i * 4;
    const int b = e >> 6;                                 // /64
    const int c = e & 63;
    *(v4f*)&As[b * 68 + c] = ((const v4f*)Ag)[i];
  }
  __syncthreads();

  const int wave = threadIdx.x >> 5;                      // 0..15
  const int lane = threadIdx.x & 31;
  const int l16  = lane & 15;                             // M (for A) / col (for B)
  const int koff = (lane >> 4) * 2;                       // K pair select (ISA 7.12.2)
  const int n_base = wave * 2;                            // this wave's 2 capsules

  const float* Wg = W + (size_t)u * (N_CAP * C_DIM * D_CAP);

  v8f acc[4][2] = {};                                     // [m-tile][n-tile]
  for (int kc = 0; kc < C_DIM; kc += 4) {
    const int k = kc + koff;
    v2f a[4], bb[2];
    // A fragment: lane l16 = row M, elements K = k, k+1 (contiguous -> b64 LDS load)
    for (int mt = 0; mt < 4; ++mt)
      a[mt] = *(const v2f*)&As[(mt * 16 + l16) * 68 + k];
    // B fragment: lane l16 = col d, VGPR j = row K=k+j. 64B-coalesced per half-wave.
    for (int nt = 0; nt < 2; ++nt) {
      const float* wp = Wg + ((size_t)(n_base + nt) * C_DIM + k) * D_CAP + l16;
      bb[nt].x = wp[0];
      bb[nt].y = wp[D_CAP];
      if (kc + 4 < C_DIM)                                 // speculative prefetch of next K chunk
        __builtin_prefetch(wp + 4 * D_CAP, 0, 1);
    }
    for (int mt = 0; mt < 4; ++mt)
      for (int nt = 0; nt < 2; ++nt)
        acc[mt][nt] = __builtin_amdgcn_wmma_f32_16x16x4_f32(
            false, a[mt], false, bb[nt], (short)0, acc[mt][nt], false, false);
  }

  // Store: tile(mt,nt) covers n = n_base+nt, d = lane&15, rows b = mt*16 + half*8 + r.
  const int d = l16;
  const int rbase = (lane >> 4) * 8;
  for (int mt = 0; mt < 4; ++mt)
    for (int nt = 0; nt < 2; ++nt) {
      const int n = n_base + nt;
      for (int r = 0; r < 8; ++r) {
        const int b = mt * 16 + rbase + r;
        uhat[(((size_t)b * N_CAP + n) * U_DIM + u) * D_CAP + d] = acc[mt][nt][r];
      }
    }
}

// ---------------------------------------------------------------------------
// Kernel 2: dynamic routing. One workgroup per (b,n); u_hat rows held in VGPRs,
// shuffle + small-LDS reductions. u_hat re-reads hit L2.
// ---------------------------------------------------------------------------
__global__ void caps_route(const float* __restrict__ uhat,  // [B][N][U][D]
                           float* __restrict__ out) {       // [B][D][N]
  const int bn = blockIdx.x;                 // b*32 + n
  const int b = bn >> 5, n = bn & 31;
  const int t = threadIdx.x;                 // 0..255
  const int lane = t & 31, wave = t >> 5;

  __shared__ float red[8 * 16];              // per-wave partial s vectors
  __shared__ float sv[16];                   // shared s vector
  __shared__ float rscal[8];                 // per-wave scalar reductions

  const float* base = uhat + (size_t)bn * (U_DIM * D_CAP);
  // Each thread owns 4 rows: u = t + k*256.
  v4f row[4][4];
  for (int k = 0; k < 4; ++k) {
    const v4f* p = (const v4f*)(base + (size_t)(t + k * 256) * D_CAP);
    for (int j = 0; j < 4; ++j) row[k][j] = p[j];
  }

  float blog[4] = {0.f, 0.f, 0.f, 0.f};
  float v[16];

  for (int it = 0; it < 3; ++it) {
    // ---- c = softmax(blog) over U ----
    float c[4];
    if (it == 0) {
      c[0] = c[1] = c[2] = c[3] = 1.0f / (float)U_DIM;   // softmax of zeros
    } else {
      float mx = fmaxf(fmaxf(blog[0], blog[1]), fmaxf(blog[2], blog[3]));
      for (int o = 16; o; o >>= 1) mx = fmaxf(mx, __shfl_xor(mx, o, 32));
      if (lane == 0) rscal[wave] = mx;
      __syncthreads();
      float m2 = rscal[0];
      for (int w = 1; w < 8; ++w) m2 = fmaxf(m2, rscal[w]);
      float sum = 0.f;
      for (int k = 0; k < 4; ++k) { c[k] = __expf(blog[k] - m2); sum += c[k]; }
      for (int o = 16; o; o >>= 1) sum += __shfl_xor(sum, o, 32);
      __syncthreads();                       // all reads of rscal(max) done
      if (lane == 0) rscal[wave] = sum;
      __syncthreads();
      float tot = 0.f;
      for (int w = 0; w < 8; ++w) tot += rscal[w];
      const float inv = 1.0f / tot;
      for (int k = 0; k < 4; ++k) c[k] *= inv;
    }

    // ---- s_d = sum_u c_u * uhat[u][d] ----
    float sp[16];
    for (int d = 0; d < 16; ++d) sp[d] = 0.f;
    for (int k = 0; k < 4; ++k)
      for (int j = 0; j < 4; ++j)
        for (int e = 0; e < 4; ++e)
          sp[j * 4 + e] += c[k] * row[k][j][e];
    for (int d = 0; d < 16; ++d)
      for (int o = 16; o; o >>= 1) sp[d] += __shfl_xor(sp[d], o, 32);
    if (lane == 0)
      for (int d = 0; d < 16; ++d) red[wave * 16 + d] = sp[d];
    __syncthreads();
    if (t < 16) {
      float s = 0.f;
      for (int w = 0; w < 8; ++w) s += red[w * 16 + t];
      sv[t] = s;
    }
    __syncthreads();

    // ---- v = squash(s) (computed redundantly by all threads) ----
    float s2 = 1e-8f;
    for (int d = 0; d < 16; ++d) s2 += sv[d] * sv[d];
    const float scale = sqrtf(s2) / (1.0f + s2);
    for (int d = 0; d < 16; ++d) v[d] = sv[d] * scale;

    // ---- blog += uhat . v ----
    for (int k = 0; k < 4; ++k) {
      float dot = 0.f;
      for (int j = 0; j < 4; ++j)
        for (int e = 0; e < 4; ++e)
          dot += row[k][j][e] * v[j * 4 + e];
      blog[k] += dot;
    }
    __syncthreads();                         // guard red/sv/rscal reuse
  }

  // out[b, d, n] = v_d
  if (t < 16)
    out[((size_t)b * D_CAP + t) * N_CAP + n] = v[t];
}

// ---------------------------------------------------------------------------
extern "C" void kernel_launch(void* const* d_in, const int* in_sizes, int n_in,
                              void* d_out, int out_size, void* d_ws, size_t ws_size,
                              hipStream_t stream) {
  const float* u_vecs = (const float*)d_in[0];   // (64, 64, 1024) f32
  const float* W      = (const float*)d_in[1];   // (1024, 32, 64, 16) f32
  float* out = (float*)d_out;                    // (64, 16, 32) f32

  // Workspace layout: uT = 16MB, u_hat = 128MB (needs ws_size >= 144MB).
  float* uT   = (float*)d_ws;                         // [U][B*C]
  float* uhat = uT + (size_t)U_DIM * B_DIM * C_DIM;   // [B][N][U][D]

  caps_transpose<<<dim3(U_DIM / 32, (B_DIM * C_DIM) / 32), 256, 0, stream>>>(u_vecs, uT);
  caps_uhat<<<dim3(U_DIM), 512, 0, stream>>>(uT, W, uhat);
  caps_route<<<dim3(B_DIM * N_CAP), 256, 0, stream>>>(uhat, out);
}